// ChannelBlockImportanceGate_22548578304113
// MI455X (gfx1250) — compile-verified
//
#include <hip/hip_runtime.h>

// ---------------- problem constants (from reference) ----------------
static constexpr int kH = 132, kW = 132;      // spatial dims
static constexpr int kB = 8;                  // BLOCK
static constexpr int kNH = 17, kNW = 17;      // ceil(132/8)
static constexpr int kTOTAL = kNH * kNW;      // 289 blocks per channel
static constexpr int kKEEP = 72;              // round(289*0.25)
static constexpr int kNT = 256;               // 8 wave32 waves per WG
static constexpr int kROWCH = kW / 4;         // 33 b128 chunks per row
static constexpr int kNVEC = (kH * kW) / 4;   // 4356 float4 stores per channel

typedef float v4f __attribute__((ext_vector_type(4)));

// ---------------- gfx1250 async global->LDS path --------------------
#if defined(__gfx1250__) && __has_builtin(__builtin_amdgcn_global_load_async_to_lds_b128)
#define USE_ASYNC_LDS 1
typedef int v4i __attribute__((vector_size(16)));
typedef __attribute__((address_space(1))) v4i* as1_v4i_ptr;   // global src
typedef __attribute__((address_space(3))) v4i* as3_v4i_ptr;   // LDS dst
#endif

__device__ __forceinline__ void wait_async_lds() {
#if defined(USE_ASYNC_LDS)
#if __has_builtin(__builtin_amdgcn_s_wait_asynccnt)
  __builtin_amdgcn_s_wait_asynccnt(0);
#else
  asm volatile("s_wait_asynccnt 0" ::: "memory");
#endif
#endif
}

__device__ __forceinline__ float sig(float x) {
  return 1.0f / (1.0f + expf(-x));
}

__global__ __launch_bounds__(kNT)
void ChannelBlockImportanceGate_kernel(const float* __restrict__ feat,
                                       const int* __restrict__ enabled,
                                       float* __restrict__ out) {
  const int tid = threadIdx.x;
  const long long ch = blockIdx.x;
  const float* __restrict__ src = feat + ch * (long long)(kH * kW);
  float* __restrict__ dst = out + ch * (long long)(kH * kW);
  v4f* __restrict__ dst4 = (v4f*)dst;  // 16B-aligned: 69696 B per channel

  __shared__ __align__(16) float panel[kB * kW];  // one 8x132 row-panel
  __shared__ float colsum[kW];
  __shared__ float scores[kTOTAL];
  __shared__ float hardf[kTOTAL];
  __shared__ float partials[kNT];
  __shared__ float s_thr;
  __shared__ float s_scale;

  if (!enabled[0]) {  // uniform branch across the whole grid
    const v4f ones = {1.0f, 1.0f, 1.0f, 1.0f};
    for (int v = tid; v < kNVEC; v += kNT)
      __builtin_nontemporal_store(ones, dst4 + v);
    return;
  }

  // ---- Phase A: per-panel abs + 8x8 block-mean -> scores[289] ----
  // Async DMA stages each 8x132 row-panel into LDS with coalesced b128
  // transfers (ASYNCcnt), then strided |x| column sums read LDS, not global.
  for (int p = 0; p < kNH; ++p) {
    const int rows = (p == kNH - 1) ? (kH - p * kB) : kB;  // 4 on last panel
    const int nchunk = rows * kROWCH;                      // b128 chunks

#if defined(USE_ASYNC_LDS)
    for (int c = tid; c < nchunk; c += kNT) {
      const int r = c / kROWCH, c4 = c - r * kROWCH;
      const float* g = src + (p * kB + r) * kW + c4 * 4;
      __builtin_amdgcn_global_load_async_to_lds_b128(
          (as1_v4i_ptr)(g),
          (as3_v4i_ptr)(&panel[r * kW + c4 * 4]),
          /*offset=*/0, /*cpol=*/0);
    }
    wait_async_lds();
#else
    for (int c = tid; c < nchunk; c += kNT) {
      const int r = c / kROWCH, c4 = c - r * kROWCH;
      const v4f v = *(const v4f*)(src + (p * kB + r) * kW + c4 * 4);
      *(v4f*)(&panel[r * kW + c4 * 4]) = v;
    }
#endif
    __syncthreads();

    // column sums of |x| over the (<=8) valid rows; pad rows are zero.
    for (int c = tid; c < kW; c += kNT) {
      float s = 0.0f;
      #pragma unroll
      for (int r = 0; r < kB; ++r)
        if (r < rows) s += fabsf(panel[r * kW + c]);
      colsum[c] = s;
    }
    __syncthreads();

    // 8-wide block sums (last block has only 4 valid cols); mean = /64 exact.
    for (int b = tid; b < kNW; b += kNT) {
      const int cmax = (b == kNW - 1) ? (kW - b * kB) : kB;
      float s = 0.0f;
      for (int k = 0; k < cmax; ++k) s += colsum[b * kB + k];
      scores[p * kNW + b] = s * (1.0f / 64.0f);
    }
    __syncthreads();
  }

  // ---- Phase B: exact top_k rank (value desc, index asc tie-break) ----
  for (int i = tid; i < kTOTAL; i += kNT) {
    const float si = scores[i];
    int rank = 0;
    for (int j = 0; j < kTOTAL; ++j) {
      const float sj = scores[j];
      rank += (sj > si) || (sj == si && j < i);
    }
    hardf[i] = (rank < kKEEP) ? 1.0f : 0.0f;
    if (rank == kKEEP - 1) s_thr = si;  // unique writer (ranks are a permutation)
  }
  __syncthreads();

  // deterministic soft-sum (tree reduction; no float LDS atomics)
  const float thr = s_thr;
  float part = 0.0f;
  for (int i = tid; i < kTOTAL; i += kNT) part += sig((scores[i] - thr) / 0.1f);
  partials[tid] = part;
  __syncthreads();
  for (int off = kNT / 2; off > 0; off >>= 1) {
    if (tid < off) partials[tid] += partials[tid + off];
    __syncthreads();
  }
  if (tid == 0) s_scale = (float)kKEEP / fmaxf(partials[0], 1e-6f);
  __syncthreads();

  // straight-through blend, same fp order as reference: (hard - soft) + soft
  const float scale = s_scale;
  for (int i = tid; i < kTOTAL; i += kNT) {
    float soft = sig((scores[i] - thr) / 0.1f) * scale;
    soft = fminf(fmaxf(soft, 0.0f), 1.0f);
    scores[i] = (hardf[i] - soft) + soft;
  }
  __syncthreads();

  // ---- Phase C: broadcast block mask to pixels ----
  // Any 4-aligned column quad lies in one 8-wide block, so each b128 store
  // needs exactly one LDS read. Non-temporal: output is write-once.
  for (int v = tid; v < kNVEC; v += kNT) {
    const int r = v / kROWCH;          // pixel row
    const int c4 = v - r * kROWCH;     // float4 column (0..32)
    const float m = scores[(r >> 3) * kNW + (c4 >> 1)];
    const v4f val = {m, m, m, m};
    __builtin_nontemporal_store(val, dst4 + v);
  }
}

extern "C" void kernel_launch(void* const* d_in, const int* in_sizes, int n_in,
                              void* d_out, int out_size, void* d_ws, size_t ws_size,
                              hipStream_t stream) {
  const float* feat = (const float*)d_in[0];
  const int* enabled = (const int*)d_in[1];
  float* out = (float*)d_out;
  const int channels = in_sizes[0] / (kH * kW);  // B*C = 2048
  hipLaunchKernelGGL(ChannelBlockImportanceGate_kernel,
                     dim3(channels), dim3(kNT), 0, stream,
                     feat, enabled, out);
}